// MAE_65025804861607
// MI455X (gfx1250) — compile-verified
//
#include <hip/hip_runtime.h>
#include <hip/hip_bf16.h>
#include <math.h>

typedef float v2f __attribute__((ext_vector_type(2)));
typedef float v8f __attribute__((ext_vector_type(8)));

#define LSP 4096   // H*W
#define CDIM 64
#define NBATCH 16

// ---------------- workspace layout (in floats) ----------------
static const size_t O_QKV = 0;                                    // 16*192*4096
static const size_t O_F72 = (size_t)16 * 192 * 4096;              // 16*72*4096
static const size_t O_XN  = O_F72 + (size_t)16 * 72 * 4096;       // 16*4096*64
static const size_t O_XG  = O_XN  + (size_t)16 * 4096 * 64;       // 64*4096*16
static const size_t O_DT  = O_XG  + (size_t)64 * 4096 * 16;       // 64*4096*16
static const size_t O_BC  = O_DT  + (size_t)64 * 4096 * 16;       // 64*4096*8
static const size_t O_CC  = O_BC  + (size_t)64 * 4096 * 8;        // 64*4096*8
static const size_t O_YS  = O_CC  + (size_t)64 * 4096 * 8;        // 64*4096*16
static const size_t O_Y   = 0;                                    // reuse qkv region (dead after f72)
static const size_t O_YP  = (size_t)16 * 4096 * 64;               // also inside old qkv region
static const size_t O_YM  = O_YS  + (size_t)64 * 4096 * 16;       // 16*64
static const size_t O_CA  = O_YM  + 1024;                         // 16*64

// ---------------- K1: qkv pointwise conv as fp32 WMMA GEMM ----------------
// qkv[b,oc,l] = sum_c qkv_w[oc,c] * x[b,c,l];  one wave -> 16(oc) x 16(l) tile
__global__ void k_qkv_gemm(const float* __restrict__ x, const float* __restrict__ w,
                           float* __restrict__ qkv) {
  int lane = threadIdx.x;
  int m = lane & 15, kh = lane >> 4;
  int lBase  = blockIdx.x * 16;
  int ocBase = blockIdx.y * 16;
  int b      = blockIdx.z;
  const float* xb = x + (size_t)b * CDIM * LSP;
  v8f acc = {0.f, 0.f, 0.f, 0.f, 0.f, 0.f, 0.f, 0.f};
  for (int k0 = 0; k0 < CDIM; k0 += 4) {
    int k = k0 + 2 * kh;
    v2f a, bb;
    a.x  = w[(ocBase + m) * CDIM + k];
    a.y  = w[(ocBase + m) * CDIM + k + 1];
    bb.x = xb[(size_t)k * LSP + lBase + m];
    bb.y = xb[(size_t)(k + 1) * LSP + lBase + m];
    acc = __builtin_amdgcn_wmma_f32_16x16x4_f32(false, a, false, bb,
                                                (short)0, acc, false, false);
  }
  float* out = qkv + ((size_t)b * 192 + ocBase) * LSP + lBase;
  for (int r = 0; r < 8; ++r) {
    int row = r + 8 * kh;
    out[(size_t)row * LSP + m] = acc[r];
  }
}

// ---------------- K2: fused depthwise 3x3 + fc mix -> f72 ----------------
// f72[b, d*9+o, l] = fc_b[o] + sum_k fc_w[o,k] * dw3x3(qkv[b, k*8+d, :])(l)
__global__ void k_f72(const float* __restrict__ qkv, const float* __restrict__ dww,
                      const float* __restrict__ fcw, const float* __restrict__ fcb,
                      float* __restrict__ f72) {
  int idx = blockIdx.x * blockDim.x + threadIdx.x;
  int l = idx & 4095; int t = idx >> 12;
  int c72 = t % 72;   int b = t / 72;
  int d = c72 / 9, o = c72 % 9;
  int h = l >> 6, w = l & 63;
  float acc = fcb[o];
  for (int k = 0; k < 24; ++k) {
    int ch = k * 8 + d;
    const float* q   = qkv + ((size_t)b * 192 + ch) * LSP;
    const float* dwp = dww + ch * 27 + 9;          // mid depth slice (kd=1)
    float s = 0.f;
    for (int kh = 0; kh < 3; ++kh) {
      int hh = h + kh - 1; if ((unsigned)hh >= 64u) continue;
      for (int kw = 0; kw < 3; ++kw) {
        int ww2 = w + kw - 1; if ((unsigned)ww2 >= 64u) continue;
        s += q[hh * 64 + ww2] * dwp[kh * 3 + kw];
      }
    }
    acc += fcw[o * 24 + k] * s;
  }
  f72[((size_t)b * 72 + c72) * LSP + l] = acc;
}

// ---------------- K3: channel LayerNorm -> xn (B,L,C) ----------------
__global__ void k_layernorm(const float* __restrict__ x, const float* __restrict__ g,
                            const float* __restrict__ bta, float* __restrict__ xn) {
  int idx = blockIdx.x * blockDim.x + threadIdx.x;  // (b,l)
  int b = idx >> 12; int l = idx & 4095;
  const float* xp = x + (size_t)b * 64 * LSP + l;
  float s = 0.f, s2 = 0.f;
  for (int c = 0; c < 64; ++c) { float v = xp[(size_t)c * LSP]; s += v; s2 += v * v; }
  float mu  = s * (1.f / 64.f);
  float var = s2 * (1.f / 64.f) - mu * mu;
  float inv = rsqrtf(var + 1e-5f);
  float* o = xn + ((size_t)b * 4096 + l) * 64;
  for (int c = 0; c < 64; ++c) {
    float v = xp[(size_t)c * LSP];
    o[c] = (v - mu) * inv * g[c] + bta[c];
  }
}

// ---------------- K4: build scan sequences + precompute dt / B / C ----------------
__global__ void k_seqbuild(const float* __restrict__ xn, const float* __restrict__ Wdt,
                           const float* __restrict__ bdt, const float* __restrict__ WB,
                           const float* __restrict__ WC, float* __restrict__ xg,
                           float* __restrict__ dto, float* __restrict__ Bc,
                           float* __restrict__ Cc) {
  int idx = blockIdx.x * blockDim.x + threadIdx.x;  // 64*4096
  int l = idx & 4095; int t = idx >> 12;
  int b = t & 15; int g = t >> 4;
  int lp = (g & 1) ? (4095 - l) : l;
  int h, w;
  if (g < 2) { h = lp >> 6; w = lp & 63; } else { w = lp >> 6; h = lp & 63; }
  int lsp = h * 64 + w;
  float xv[16];
  const float* src = xn + ((size_t)b * 4096 + lsp) * 64 + g * 16;
  for (int d = 0; d < 16; ++d) xv[d] = src[d];
  size_t base16 = ((size_t)t * 4096 + l) * 16;
  size_t base8  = ((size_t)t * 4096 + l) * 8;
  for (int d = 0; d < 16; ++d) xg[base16 + d] = xv[d];
  const float* Wg = Wdt + g * 256;
  for (int e = 0; e < 16; ++e) {
    float s = bdt[g * 16 + e];
    for (int d = 0; d < 16; ++d) s += xv[d] * Wg[d * 16 + e];
    dto[base16 + e] = (s > 20.f) ? s : log1pf(__expf(s));   // softplus
  }
  for (int n = 0; n < 8; ++n) {
    float sb = 0.f, sc = 0.f;
    for (int d = 0; d < 16; ++d) {
      sb += xv[d] * WB[g * 128 + d * 8 + n];
      sc += xv[d] * WC[g * 128 + d * 8 + n];
    }
    Bc[base8 + n] = sb; Cc[base8 + n] = sc;
  }
}

// ---------------- K5: sequential selective scan ----------------
// block per (g,b); 128 threads, tid = d*8+n owns one state element
__global__ void k_scan(const float* __restrict__ A_log, const float* __restrict__ Dp,
                       const float* __restrict__ xg, const float* __restrict__ dti,
                       const float* __restrict__ Bc, const float* __restrict__ Cc,
                       float* __restrict__ ys) {
  int gb = blockIdx.x;  int g = gb >> 4;
  int tid = threadIdx.x;
  int d = tid >> 3, n = tid & 7;
  float A  = -__expf(A_log[g * 128 + d * 8 + n]);
  float dp = Dp[g * 16 + d];
  const float* dtp = dti + (size_t)gb * 4096 * 16;
  const float* xp  = xg  + (size_t)gb * 4096 * 16;
  const float* bp  = Bc  + (size_t)gb * 4096 * 8;
  const float* cp  = Cc  + (size_t)gb * 4096 * 8;
  float* yp = ys + (size_t)gb * 4096 * 16;
  float h = 0.f;
  for (int l = 0; l < 4096; ++l) {
    if ((l & 15) == 0 && l + 32 < 4096)
      __builtin_prefetch(&dtp[(l + 32) * 16 + d], 0, 1);   // global_prefetch_b8
    float dt = dtp[l * 16 + d];
    float xv = xp[l * 16 + d];
    float bv = bp[l * 8 + n];
    float cv = cp[l * 8 + n];
    h = __expf(dt * A) * h + dt * bv * xv;
    float p = h * cv;
    p += __shfl_xor(p, 1, 32);
    p += __shfl_xor(p, 2, 32);
    p += __shfl_xor(p, 4, 32);
    if (n == 0) yp[l * 16 + d] = p + dp * xv;
  }
}

// ---------------- K6: undo flip/transpose, concat groups -> y (B,L,C) ----------------
__global__ void k_reorder(const float* __restrict__ ys, float* __restrict__ y) {
  int idx = blockIdx.x * blockDim.x + threadIdx.x;  // 64*4096
  int l = idx & 4095; int t = idx >> 12;
  int b = t & 15; int g = t >> 4;
  int lp = (g & 1) ? (4095 - l) : l;
  int h, w;
  if (g < 2) { h = lp >> 6; w = lp & 63; } else { w = lp >> 6; h = lp & 63; }
  int lsp = h * 64 + w;
  const float* src = ys + ((size_t)t * 4096 + l) * 16;
  float* dst = y + ((size_t)b * 4096 + lsp) * 64 + g * 16;
  for (int d = 0; d < 16; ++d) dst[d] = src[d];
}

// ---------------- K7: output projection GEMM (fp32 WMMA) ----------------
// yp[m,oc] = sum_c y[m,c]*out_w[oc,c] + out_b[oc]; m over B*L rows
__global__ void k_proj_gemm(const float* __restrict__ y, const float* __restrict__ ow,
                            const float* __restrict__ ob, float* __restrict__ yp) {
  int lane = threadIdx.x;
  int m = lane & 15, kh = lane >> 4;
  int mBase = blockIdx.x * 16;
  int nBase = blockIdx.y * 16;
  v8f acc = {0.f, 0.f, 0.f, 0.f, 0.f, 0.f, 0.f, 0.f};
  for (int k0 = 0; k0 < 64; k0 += 4) {
    int k = k0 + 2 * kh;
    v2f a, bb;
    a.x  = y[(size_t)(mBase + m) * 64 + k];
    a.y  = y[(size_t)(mBase + m) * 64 + k + 1];
    bb.x = ow[(nBase + m) * 64 + k];
    bb.y = ow[(nBase + m) * 64 + k + 1];
    acc = __builtin_amdgcn_wmma_f32_16x16x4_f32(false, a, false, bb,
                                                (short)0, acc, false, false);
  }
  for (int r = 0; r < 8; ++r) {
    int row = r + 8 * kh;
    yp[(size_t)(mBase + row) * 64 + nBase + m] = acc[r] + ob[nBase + m];
  }
}

// ---------------- K8: mean over L ----------------
__global__ void k_mean(const float* __restrict__ yp, float* __restrict__ ym) {
  int idx = blockIdx.x * blockDim.x + threadIdx.x;  // 16*64
  int b = idx >> 6, c = idx & 63;
  const float* p = yp + (size_t)b * 4096 * 64 + c;
  float s = 0.f;
  for (int l = 0; l < 4096; ++l) s += p[(size_t)l * 64];
  ym[idx] = s * (1.f / 4096.f);
}

// ---------------- K9: channel attention ----------------
__global__ void k_ca(const float* __restrict__ ym, const float* __restrict__ w1,
                     const float* __restrict__ b1, const float* __restrict__ w2,
                     const float* __restrict__ b2, float* __restrict__ ca) {
  __shared__ float hid[16];
  int b = blockIdx.x, t = threadIdx.x;
  if (t < 16) {
    float s = b1[t];
    for (int c = 0; c < 64; ++c) s += ym[b * 64 + c] * w1[t * 64 + c];
    hid[t] = fmaxf(s, 0.f);
  }
  __syncthreads();
  float s = b2[t];
  for (int j = 0; j < 16; ++j) s += hid[j] * w2[t * 16 + j];
  ca[b * 64 + t] = 1.f / (1.f + __expf(-s));
}

// ---------------- K10: grouped 3x3 conv + residual + gated mamba -> out ----------------
__global__ void k_final(const float* __restrict__ x, const float* __restrict__ f72,
                        const float* __restrict__ depw, const float* __restrict__ depb,
                        const float* __restrict__ yp, const float* __restrict__ ca,
                        float* __restrict__ out) {
  int idx = blockIdx.x * blockDim.x + threadIdx.x;  // 16*64*4096
  int l = idx & 4095; int t = idx >> 12;
  int oc = t & 63; int b = t >> 6;
  int h = l >> 6, w = l & 63;
  int grp = oc >> 3;
  float conv = depb[oc];
  for (int i9 = 0; i9 < 9; ++i9) {
    const float* fp = f72 + ((size_t)b * 72 + grp * 9 + i9) * LSP;
    const float* wp = depw + oc * 243 + i9 * 27 + 9;     // mid depth slice
    for (int kh = 0; kh < 3; ++kh) {
      int hh = h + kh - 1; if ((unsigned)hh >= 64u) continue;
      for (int kw = 0; kw < 3; ++kw) {
        int ww2 = w + kw - 1; if ((unsigned)ww2 >= 64u) continue;
        conv += fp[hh * 64 + ww2] * wp[kh * 3 + kw];
      }
    }
  }
  float mam = yp[((size_t)b * 4096 + l) * 64 + oc] * ca[b * 64 + oc];
  size_t xi = ((size_t)b * 64 + oc) * LSP + l;
  out[xi] = x[xi] + mam + conv;
}

extern "C" void kernel_launch(void* const* d_in, const int* in_sizes, int n_in,
                              void* d_out, int out_size, void* d_ws, size_t ws_size,
                              hipStream_t stream) {
  (void)in_sizes; (void)n_in; (void)out_size; (void)ws_size;
  const float* x     = (const float*)d_in[0];
  const float* qkv_w = (const float*)d_in[1];
  const float* dw_w  = (const float*)d_in[2];
  const float* fc_w  = (const float*)d_in[3];
  const float* fc_b  = (const float*)d_in[4];
  const float* dep_w = (const float*)d_in[5];
  const float* dep_b = (const float*)d_in[6];
  const float* ln_g  = (const float*)d_in[7];
  const float* ln_b  = (const float*)d_in[8];
  const float* A_log = (const float*)d_in[9];
  const float* Wdt   = (const float*)d_in[10];
  const float* bdt   = (const float*)d_in[11];
  const float* WB    = (const float*)d_in[12];
  const float* WC    = (const float*)d_in[13];
  const float* Dp    = (const float*)d_in[14];
  const float* out_w = (const float*)d_in[15];
  const float* out_b = (const float*)d_in[16];
  const float* ca_w1 = (const float*)d_in[17];
  const float* ca_b1 = (const float*)d_in[18];
  const float* ca_w2 = (const float*)d_in[19];
  const float* ca_b2 = (const float*)d_in[20];
  float* out = (float*)d_out;
  float* ws  = (float*)d_ws;

  float* qkv = ws + O_QKV;
  float* f72 = ws + O_F72;
  float* xn  = ws + O_XN;
  float* xg  = ws + O_XG;
  float* dt  = ws + O_DT;
  float* Bc  = ws + O_BC;
  float* Cc  = ws + O_CC;
  float* ys  = ws + O_YS;
  float* y   = ws + O_Y;    // reuses qkv region (qkv dead after k_f72)
  float* yp  = ws + O_YP;   // also inside old qkv region
  float* ym  = ws + O_YM;
  float* ca  = ws + O_CA;

  // attention branch
  k_qkv_gemm<<<dim3(256, 12, 16), 32, 0, stream>>>(x, qkv_w, qkv);
  k_f72<<<(16 * 72 * 4096) / 256, 256, 0, stream>>>(qkv, dw_w, fc_w, fc_b, f72);

  // mamba branch
  k_layernorm<<<(16 * 4096) / 256, 256, 0, stream>>>(x, ln_g, ln_b, xn);
  k_seqbuild<<<(64 * 4096) / 256, 256, 0, stream>>>(xn, Wdt, bdt, WB, WC, xg, dt, Bc, Cc);
  k_scan<<<64, 128, 0, stream>>>(A_log, Dp, xg, dt, Bc, Cc, ys);
  k_reorder<<<(64 * 4096) / 256, 256, 0, stream>>>(ys, y);
  k_proj_gemm<<<dim3(4096, 4), 32, 0, stream>>>(y, out_w, out_b, yp);
  k_mean<<<4, 256, 0, stream>>>(yp, ym);
  k_ca<<<16, 64, 0, stream>>>(ym, ca_w1, ca_b1, ca_w2, ca_b2, ca);

  // fuse: residual + grouped conv + gated mamba
  k_final<<<(16 * 64 * 4096) / 256, 256, 0, stream>>>(x, f72, dep_w, dep_b, yp, ca, out);
}